// JaggedArgmaxModule_49314814492716
// MI455X (gfx1250) — compile-verified
//
#include <hip/hip_runtime.h>

#define WAVE_SZ      32
#define N_WAVES      8
#define BLOCK_SZ     (N_WAVES * WAVE_SZ)

// ---- CDNA5 async global->LDS copy helpers (gfx1250) -------------------------

__device__ __forceinline__ void asyncLoadF32(const float* g, float* l) {
#if __has_builtin(__builtin_amdgcn_global_load_async_to_lds_b32)
    __builtin_amdgcn_global_load_async_to_lds_b32(
        (__attribute__((address_space(1))) int*)(g),
        (__attribute__((address_space(3))) int*)(l),
        /*offset=*/0, /*cpol=*/0);
#else
    // GV mode: vdst = 32-bit LDS byte address, vaddr = 64-bit global address
    unsigned loff = (unsigned)(unsigned long long)
        (__attribute__((address_space(3))) float*)l;
    asm volatile("global_load_async_to_lds_b32 %0, %1, off"
                 :
                 : "v"(loff), "v"(g)
                 : "memory");
#endif
}

__device__ __forceinline__ void waitAsync0() {
#if __has_builtin(__builtin_amdgcn_s_wait_asynccnt)
    __builtin_amdgcn_s_wait_asynccnt(0);
#else
    asm volatile("s_wait_asynccnt 0x0" ::: "memory");
#endif
    asm volatile("" ::: "memory");
}

__device__ __forceinline__ void waitAsync1() {
#if __has_builtin(__builtin_amdgcn_s_wait_asynccnt)
    __builtin_amdgcn_s_wait_asynccnt(1);
#else
    asm volatile("s_wait_asynccnt 0x1" ::: "memory");
#endif
    asm volatile("" ::: "memory");
}

// ---- Kernel: one wave32 per jagged segment ---------------------------------

__global__ __launch_bounds__(BLOCK_SZ)
void jagged_argmax_kernel(const float* __restrict__ values,
                          const long long* __restrict__ prefix_sum,
                          int* __restrict__ out,
                          int n_seg)
{
    // per-wave double buffer: 2 x 32 floats
    __shared__ float sbuf[N_WAVES * 2 * WAVE_SZ];

    const int lane = threadIdx.x & (WAVE_SZ - 1);
    const int wave = threadIdx.x >> 5;
    const int seg  = blockIdx.x * N_WAVES + wave;
    if (seg >= n_seg) return;

    const int end   = (int)prefix_sum[seg];
    const int start = (seg == 0) ? 0 : (int)prefix_sum[seg - 1];

    float* mybuf = &sbuf[wave * (2 * WAVE_SZ)];

    float bestV = -__builtin_inff();
    int   bestI = 0x7fffffff;   // segment_min identity for empty segments

    const int len    = end - start;
    const int nChunk = (len + WAVE_SZ - 1) >> 5;

    if (nChunk > 0) {
        // prime the pipeline: chunk 0 -> buffer 0
        {
            int idx = start + lane;
            if (idx < end) asyncLoadF32(values + idx, &mybuf[lane]);
        }
        for (int c = 0; c < nChunk; ++c) {
            if (c + 1 < nChunk) {
                // issue chunk c+1 into the other half, then wait for chunk c
                int idx = start + (c + 1) * WAVE_SZ + lane;
                if (idx < end)
                    asyncLoadF32(values + idx,
                                 &mybuf[((c + 1) & 1) * WAVE_SZ + lane]);
                waitAsync1();   // in-order completion => chunk c has landed
            } else {
                waitAsync0();   // drain the last chunk
            }
            int idx = start + c * WAVE_SZ + lane;
            float v = mybuf[(c & 1) * WAVE_SZ + lane];
            if (idx < end && v > bestV) {   // strict '>' keeps first occurrence
                bestV = v;
                bestI = idx;
            }
        }
    }

    // wave32 cross-lane argmax reduction (first-index tiebreak)
#pragma unroll
    for (int off = 16; off > 0; off >>= 1) {
        float ov = __shfl_xor(bestV, off, WAVE_SZ);
        int   oi = __shfl_xor(bestI, off, WAVE_SZ);
        if (ov > bestV || (ov == bestV && oi < bestI)) {
            bestV = ov;
            bestI = oi;
        }
    }

    if (lane == 0) out[seg] = bestI;
}

// ---- Launch ----------------------------------------------------------------

extern "C" void kernel_launch(void* const* d_in, const int* in_sizes, int n_in,
                              void* d_out, int out_size, void* d_ws, size_t ws_size,
                              hipStream_t stream)
{
    const float*     values = (const float*)d_in[0];
    const long long* prefix = (const long long*)d_in[1];
    int*             out    = (int*)d_out;   // reference output dtype is int32

    const int n_seg  = in_sizes[1];
    const int blocks = (n_seg + N_WAVES - 1) / N_WAVES;

    jagged_argmax_kernel<<<blocks, BLOCK_SZ, 0, stream>>>(values, prefix, out, n_seg);
}